// pose3d_flight_73830487818418
// MI455X (gfx1250) — compile-verified
//
#include <hip/hip_runtime.h>

typedef __attribute__((ext_vector_type(2))) float v2f;
typedef __attribute__((ext_vector_type(8))) float v8f;

#define NJ 21
#define NB 20
#define WFRAMES 131072
#define WP1 131073
#define BLOCK 128
#define NBLOCKS (WFRAMES / BLOCK)   /* 1024, multiple of 64 */
#define ROWS (BLOCK + 2)            /* window f0-2 .. f0+BLOCK-1 */
#define FSTRIDE (3 * NJ)            /* 63 floats per pose3d frame */

// ---------------------------------------------------------------------------
// Kernel 1: one thread per frame; pose3d window staged through LDS so each
// frame is read from HBM exactly once (smooth term reuses f-1, f-2 from LDS).
// Deterministic LDS tree reduction -> one fp32 partial per block.
// ---------------------------------------------------------------------------
__global__ __launch_bounds__(BLOCK) void pose_loss_kernel(
    const float* __restrict__ pose3d,   // (W+1, 3, 21)
    const float* __restrict__ bone2d,   // (W, 2, 21)
    const float* __restrict__ Rd,       // (W, 3, 3)
    const float* __restrict__ Cd,       // (W, 3, 1)
    const float* __restrict__ lift,     // (W, 3, 20)
    const float* __restrict__ blen,     // (20,)
    const int*   __restrict__ bidx,     // (20, 2)
    const int*   __restrict__ lidx,     // (20, 2)
    float* __restrict__ partial)        // (NBLOCKS,)
{
    __shared__ float sm[ROWS * FSTRIDE];   // 130*63*4 = 32760 B
    __shared__ float red[BLOCK];

    const int tid = threadIdx.x;
    const int f0  = blockIdx.x * BLOCK;

    // ---- stage pose3d rows [f0-2 .. f0+BLOCK-1] (mod WP1) into LDS ----
    if (f0 >= 2) {
        // contiguous window: flat coalesced copy
        const float* src = pose3d + (size_t)(f0 - 2) * FSTRIDE;
        for (int i = tid; i < ROWS * FSTRIDE; i += BLOCK) sm[i] = src[i];
    } else {
        // block 0: rows -2,-1 wrap to WP1-2, WP1-1
        for (int r = tid; r < ROWS; r += BLOCK) {
            int g = f0 - 2 + r;
            if (g < 0) g += WP1;
            const float* src = pose3d + (size_t)g * FSTRIDE;
            float* dst = sm + r * FSTRIDE;
            for (int i = 0; i < FSTRIDE; ++i) dst[i] = src[i];
        }
    }
    __syncthreads();

    const int f = f0 + tid;
    const float* P  = sm + (tid + 2) * FSTRIDE;  // frame f
    const float* p1 = sm + (tid + 1) * FSTRIDE;  // frame (f-1) mod WP1
    const float* p2 = sm + (tid + 0) * FSTRIDE;  // frame (f-2) mod WP1

    // ---- smooth: sum (P - 2*p1 + p2)^2 over 63 elements ----
    float sacc = 0.f;
    for (int i = 0; i < FSTRIDE; ++i) {
        float d = P[i] - 2.f * p1[i] + p2[i];
        sacc += d * d;
    }

    // ---- projection: Pc = R^T (P - C), pinhole, MSE vs bone2d ----
    const float* Rf = Rd + (size_t)f * 9;   // R[j][i] at j*3+i
    const float* Cf = Cd + (size_t)f * 3;
    const float r00 = Rf[0], r01 = Rf[1], r02 = Rf[2];
    const float r10 = Rf[3], r11 = Rf[4], r12 = Rf[5];
    const float r20 = Rf[6], r21 = Rf[7], r22 = Rf[8];
    const float c0 = Cf[0], c1 = Cf[1], c2 = Cf[2];
    const float* b2 = bone2d + (size_t)f * (2 * NJ);

    float pacc = 0.f;
    for (int k = 0; k < NJ; ++k) {
        float px = P[0 * NJ + k] - c0;
        float py = P[1 * NJ + k] - c1;
        float pz = P[2 * NJ + k] - c2;
        float pc0 = r00 * px + r10 * py + r20 * pz;   // i = 0
        float pc1 = r01 * px + r11 * py + r21 * pz;   // i = 1
        float pc2 = r02 * px + r12 * py + r22 * pz;   // i = 2  (= z)
        float inv = 1.f / pc2;
        float e0 = 512.f * pc0 * inv + 512.f - b2[k];
        float e1 = 512.f * pc1 * inv + 512.f - b2[NJ + k];
        pacc += e0 * e0 + e1 * e1;
    }

    // ---- lift (bone directions) + bone length terms ----
    const float* lf = lift + (size_t)f * (3 * NB);
    float lacc = 0.f, bacc = 0.f;
    for (int b = 0; b < NB; ++b) {
        int li0 = lidx[2 * b], li1 = lidx[2 * b + 1];
        float v0 = P[0 * NJ + li0] - P[0 * NJ + li1];
        float v1 = P[1 * NJ + li0] - P[1 * NJ + li1];
        float v2 = P[2 * NJ + li0] - P[2 * NJ + li1];
        float inv = 1.f / (sqrtf(v0 * v0 + v1 * v1 + v2 * v2) + 1e-10f);
        float d0 = lf[0 * NB + b] - v0 * inv;
        float d1 = lf[1 * NB + b] - v1 * inv;
        float d2 = lf[2 * NB + b] - v2 * inv;
        lacc += d0 * d0 + d1 * d1 + d2 * d2;

        int bi0 = bidx[2 * b], bi1 = bidx[2 * b + 1];
        float u0 = P[0 * NJ + bi0] - P[0 * NJ + bi1];
        float u1 = P[1 * NJ + bi0] - P[1 * NJ + bi1];
        float u2 = P[2 * NJ + bi0] - P[2 * NJ + bi1];
        float l2 = u0 * u0 + u1 * u1 + u2 * u2;
        float e = blen[b] - l2;
        bacc += e * e;
    }

    const float mask = (f != 0) ? 1.f : 0.f;
    float loss = 0.3f * (pacc * (1.f / 42.f)) * mask
               + 0.5f * (sacc * (1.f / 63.f))
               + 0.2f * (bacc * (1.f / 20.f))
               + 0.1f * (lacc * (1.f / 60.f)) * mask;

    // ---- deterministic block tree reduction ----
    red[tid] = loss;
    __syncthreads();
    for (int s = BLOCK / 2; s > 0; s >>= 1) {
        if (tid < s) red[tid] += red[tid + s];
        __syncthreads();
    }
    if (tid == 0) partial[blockIdx.x] = red[0];
}

// ---------------------------------------------------------------------------
// Kernel 2: single wave32 reduces the 1024 block partials with
// V_WMMA_F32_16X16X4_F32 against an all-ones B matrix:
//   D[i][j] = sum_k A[i][k] + C[i][j]  ->  sum_i D[i][0] = total of A.
// Since B is all-ones and only the total is needed, A/B VGPR layout is
// irrelevant: any placement of the 64 partials per iteration is exact.
// Column N=0 of D lives in lanes 0 (M=0..7) and 16 (M=8..15).
// ---------------------------------------------------------------------------
__global__ __launch_bounds__(32) void reduce_wmma_kernel(
    const float* __restrict__ partial, float* __restrict__ out)
{
    const int lane = threadIdx.x;
    v8f acc = {0.f, 0.f, 0.f, 0.f, 0.f, 0.f, 0.f, 0.f};
    v2f ones;
    ones.x = 1.f;
    ones.y = 1.f;

    for (int base = 0; base < NBLOCKS; base += 64) {
        v2f a;
        a.x = partial[base + lane];
        a.y = partial[base + 32 + lane];
        acc = __builtin_amdgcn_wmma_f32_16x16x4_f32(
            /*neg_a=*/false, a, /*neg_b=*/false, ones,
            /*c_mod=*/(short)0, acc, /*reuse_a=*/false, /*reuse_b=*/false);
    }

    float t = acc[0] + acc[1] + acc[2] + acc[3] +
              acc[4] + acc[5] + acc[6] + acc[7];   // lane0: D[0..7][0]; lane16: D[8..15][0]
    float o = __shfl(t, 16, 32);
    if (lane == 0) out[0] = t + o;
}

extern "C" void kernel_launch(void* const* d_in, const int* in_sizes, int n_in,
                              void* d_out, int out_size, void* d_ws, size_t ws_size,
                              hipStream_t stream) {
    const float* pose3d = (const float*)d_in[0];
    const float* bone2d = (const float*)d_in[1];
    const float* Rd     = (const float*)d_in[2];
    const float* Cd     = (const float*)d_in[3];
    const float* lift   = (const float*)d_in[4];
    const float* blen   = (const float*)d_in[5];
    const int*   bidx   = (const int*)d_in[6];
    const int*   lidx   = (const int*)d_in[7];
    float* partial = (float*)d_ws;   // NBLOCKS floats, fully overwritten each call
    float* out = (float*)d_out;

    pose_loss_kernel<<<NBLOCKS, BLOCK, 0, stream>>>(
        pose3d, bone2d, Rd, Cd, lift, blen, bidx, lidx, partial);
    reduce_wmma_kernel<<<1, 32, 0, stream>>>(partial, out);
}